// EuclideanSimilarity_21835613733206
// MI455X (gfx1250) — compile-verified
//
#include <hip/hip_runtime.h>
#include <hip/hip_bf16.h>

typedef __attribute__((ext_vector_type(2))) float v2f;
typedef __attribute__((ext_vector_type(4))) float v4f;
typedef __attribute__((ext_vector_type(8))) float v8f;

#define DIMD 128
#define NROWS 8192   // N (z_anc rows)
#define MROWS 8192   // M (z_pos_neg rows)

// ---------------------------------------------------------------------------
// Kernel 1: squared row norms for both matrices into workspace.
// 16384 threads, one row each, float4 streaming loads (8 MB total -> trivial).
// ---------------------------------------------------------------------------
__global__ __launch_bounds__(256)
void euclid_norms_kernel(const float* __restrict__ a,
                         const float* __restrict__ b,
                         float* __restrict__ a2,
                         float* __restrict__ b2) {
    int r = blockIdx.x * blockDim.x + threadIdx.x;   // 0 .. 16383
    const float* src = (r < NROWS) ? (a + (size_t)r * DIMD)
                                   : (b + (size_t)(r - NROWS) * DIMD);
    float s = 0.0f;
#pragma unroll
    for (int i = 0; i < DIMD; i += 4) {
        v4f v = *(const v4f*)(src + i);
        s = fmaf(v.x, v.x, s);
        s = fmaf(v.y, v.y, s);
        s = fmaf(v.z, v.z, s);
        s = fmaf(v.w, v.w, s);
    }
    if (r < NROWS) a2[r] = s;
    else           b2[r - NROWS] = s;
}

// ---------------------------------------------------------------------------
// Kernel 2: -||a_i - b_j|| via f32 WMMA GEMM.
// Block = 256 threads = 8 waves. Wave w computes rows [ (by*8+w)*16 , +16 )
// and cols [ bx*64 , +64 ) -> 4 accumulators of 16x16.
// K-loop: 32 steps of V_WMMA_F32_16X16X4_F32.
// ---------------------------------------------------------------------------
__global__ __launch_bounds__(256)
void euclid_wmma_kernel(const float* __restrict__ A,   // [8192,128] z_anc
                        const float* __restrict__ B,   // [8192,128] z_pos_neg
                        const float* __restrict__ a2,  // [8192]
                        const float* __restrict__ b2,  // [8192]
                        float* __restrict__ out) {     // [8192,8192]
    const int lane = threadIdx.x & 31;
    const int wv   = threadIdx.x >> 5;

    const int m0 = (blockIdx.y * 8 + wv) * 16;   // row tile start
    const int n0 = blockIdx.x * 64;              // col strip start

    const int half = lane >> 4;   // 0: K pair {0,1}; 1: K pair {2,3}
    const int l16  = lane & 15;

    // A fragment source: row m0+l16, starting at K offset 2*half.
    const float* arow  = A + (size_t)(m0 + l16) * DIMD + 2 * half;
    // B fragments (row-major z_pos_neg == B^T): rows n0+16t+l16.
    const float* brow0 = B + (size_t)(n0 +  0 + l16) * DIMD + 2 * half;
    const float* brow1 = B + (size_t)(n0 + 16 + l16) * DIMD + 2 * half;
    const float* brow2 = B + (size_t)(n0 + 32 + l16) * DIMD + 2 * half;
    const float* brow3 = B + (size_t)(n0 + 48 + l16) * DIMD + 2 * half;

    v8f acc0 = {}, acc1 = {}, acc2 = {}, acc3 = {};

#pragma unroll 8
    for (int k = 0; k < DIMD; k += 4) {
        v2f af  = *(const v2f*)(arow  + k);
        v2f bf0 = *(const v2f*)(brow0 + k);
        v2f bf1 = *(const v2f*)(brow1 + k);
        v2f bf2 = *(const v2f*)(brow2 + k);
        v2f bf3 = *(const v2f*)(brow3 + k);
        // (neg_a, A, neg_b, B, c_mod, C, reuse_a, reuse_b)
        acc0 = __builtin_amdgcn_wmma_f32_16x16x4_f32(false, af, false, bf0,
                                                     (short)0, acc0, false, false);
        acc1 = __builtin_amdgcn_wmma_f32_16x16x4_f32(false, af, false, bf1,
                                                     (short)0, acc1, false, false);
        acc2 = __builtin_amdgcn_wmma_f32_16x16x4_f32(false, af, false, bf2,
                                                     (short)0, acc2, false, false);
        acc3 = __builtin_amdgcn_wmma_f32_16x16x4_f32(false, af, false, bf3,
                                                     (short)0, acc3, false, false);
    }

    // C/D layout: VGPR i -> M = i + 8*half ; lane l16 -> N.
    float a2v[8];
#pragma unroll
    for (int i = 0; i < 8; ++i)
        a2v[i] = a2[m0 + i + 8 * half];

    float accs[4][8];
#pragma unroll
    for (int i = 0; i < 8; ++i) { accs[0][i] = acc0[i]; accs[1][i] = acc1[i];
                                  accs[2][i] = acc2[i]; accs[3][i] = acc3[i]; }

#pragma unroll
    for (int t = 0; t < 4; ++t) {
        const int col = n0 + 16 * t + l16;
        const float b2v = b2[col];
#pragma unroll
        for (int i = 0; i < 8; ++i) {
            const int row = m0 + i + 8 * half;
            float d2 = a2v[i] + b2v - 2.0f * accs[t][i];
            d2 = fmaxf(d2, 0.0f);
            out[(size_t)row * MROWS + col] = -__builtin_sqrtf(d2);
        }
    }
}

// ---------------------------------------------------------------------------
extern "C" void kernel_launch(void* const* d_in, const int* in_sizes, int n_in,
                              void* d_out, int out_size, void* d_ws, size_t ws_size,
                              hipStream_t stream) {
    const float* z_anc     = (const float*)d_in[0];   // [8192,128]
    const float* z_pos_neg = (const float*)d_in[1];   // [8192,128]
    float* out = (float*)d_out;                       // [8192,8192]

    float* a2 = (float*)d_ws;            // 8192 floats
    float* b2 = a2 + NROWS;              // 8192 floats  (64 KB total)

    // 1) row norms
    euclid_norms_kernel<<<dim3((NROWS + MROWS) / 256), dim3(256), 0, stream>>>(
        z_anc, z_pos_neg, a2, b2);

    // 2) WMMA GEMM + distance epilogue
    //    grid.x: 8192/64 = 128 column strips; grid.y: 8192/(16*8) = 64 row groups
    euclid_wmma_kernel<<<dim3(128, 64), dim3(256), 0, stream>>>(
        z_anc, z_pos_neg, a2, b2, out);
}